// GAT1_1958505087516
// MI455X (gfx1250) — compile-verified
//
#include <hip/hip_runtime.h>
#include <math.h>

// ---------------------------------------------------------------------------
// 2-layer single-head GATConv for MI455X (gfx1250, wave32).
// GEMMs use V_WMMA_F32_16X16X4_F32 (exact f32 matrix path on CDNA5);
// one wave computes a 16x64 output strip (4 accumulators) so each A
// fragment is reused by 4 WMMAs. Edge softmax/aggregation via float
// atomics on an L2-resident working set; gathers are b128-vectorized.
// ---------------------------------------------------------------------------

#define DIN  256
#define HID  128
#define OUTC 64
#define NEG_SLOPE 0.2f
#define GAT_EPS 1e-16f

typedef __attribute__((ext_vector_type(2))) float v2f;
typedef __attribute__((ext_vector_type(8))) float v8f;

// ---------------- WMMA GEMM: C[N x CC] = A[N x K] @ W[K x CC] --------------
// One wave per 16(row) x 64(col) strip: 4 x v8f accumulators.
// Block = 128 threads = 4 waves, each wave a different 16-row slab.
// Grid: x = ceil(N/64), y = CC/64.  N % 16 == 0, CC % 64 == 0 required.
// f32 A 16x4 layout (ISA 7.12.2): lanes 0-15 -> M=0..15, VGPR0/1 hold
// K{0,1} (lanes 0-15) / K{2,3} (lanes 16-31); B/D rows striped across lanes.
__global__ void gemm_wmma_f32(const float* __restrict__ A,
                              const float* __restrict__ W,
                              float* __restrict__ C,
                              int N, int K, int CC) {
  const int wave = threadIdx.x >> 5;                 // 0..3
  const int lane = threadIdx.x & 31;
  const int half = lane >> 4;                        // K pair select
  const int l16  = lane & 15;
  const int rowBase = (blockIdx.x * 4 + wave) * 16;
  const int colBase = blockIdx.y * 64;
  if (rowBase >= N) return;                          // wave-uniform exit

  const float* __restrict__ arow = A + (size_t)(rowBase + l16) * K + half * 2;
  const float* __restrict__ wcol = W + colBase + l16;

  v8f acc0 = {}, acc1 = {}, acc2 = {}, acc3 = {};
  for (int k0 = 0; k0 < K; k0 += 4) {
    v2f a;
    a.x = arow[k0];
    a.y = arow[k0 + 1];
    const size_t r0 = (size_t)(k0 + half * 2) * CC;
    const size_t r1 = r0 + CC;
    v2f b0, b1, b2, b3;
    b0.x = wcol[r0 +  0]; b0.y = wcol[r1 +  0];
    b1.x = wcol[r0 + 16]; b1.y = wcol[r1 + 16];
    b2.x = wcol[r0 + 32]; b2.y = wcol[r1 + 32];
    b3.x = wcol[r0 + 48]; b3.y = wcol[r1 + 48];
    acc0 = __builtin_amdgcn_wmma_f32_16x16x4_f32(false, a, false, b0, (short)0, acc0, false, false);
    acc1 = __builtin_amdgcn_wmma_f32_16x16x4_f32(false, a, false, b1, (short)0, acc1, false, false);
    acc2 = __builtin_amdgcn_wmma_f32_16x16x4_f32(false, a, false, b2, (short)0, acc2, false, false);
    acc3 = __builtin_amdgcn_wmma_f32_16x16x4_f32(false, a, false, b3, (short)0, acc3, false, false);
  }
  // D: VGPR r -> row rowBase + half*8 + r, col colBase + l16 (+16*tile)
  float* __restrict__ crow = C + (size_t)(rowBase + half * 8) * CC + colBase + l16;
#pragma unroll
  for (int r = 0; r < 8; ++r) {
    crow[(size_t)r * CC +  0] = acc0[r];
    crow[(size_t)r * CC + 16] = acc1[r];
    crow[(size_t)r * CC + 32] = acc2[r];
    crow[(size_t)r * CC + 48] = acc3[r];
  }
}

// -------- fold attention vectors through weights: o[k] = sum_c W[k,c]*att[c]
__global__ void wvec2(const float* __restrict__ W1, const float* __restrict__ att1,
                      const float* __restrict__ W2, const float* __restrict__ att2,
                      float* __restrict__ o1, float* __restrict__ o2,
                      int K, int C) {
  int k = blockIdx.x * blockDim.x + threadIdx.x;
  if (k >= K) return;
  float s1 = 0.f, s2 = 0.f;
  for (int c = 0; c < C; ++c) {
    s1 += W1[(size_t)k * C + c] * att1[c];
    s2 += W2[(size_t)k * C + c] * att2[c];
  }
  o1[k] = s1;
  o2[k] = s2;
}

// -------- per-row dual dot: a1[n] = X[n,:].w1 ; a2[n] = X[n,:].w2 ----------
__global__ void rowdot2(const float* __restrict__ X,
                        const float* __restrict__ w1, const float* __restrict__ w2,
                        float* __restrict__ o1, float* __restrict__ o2,
                        int N, int K) {
  int row  = blockIdx.x * (blockDim.x >> 5) + (threadIdx.x >> 5);
  int lane = threadIdx.x & 31;
  if (row >= N) return;
  const float* __restrict__ xr = X + (size_t)row * K;
  float s1 = 0.f, s2 = 0.f;
  for (int c = lane * 4; c < K; c += 128) {           // K % 128 == 0 here
    float4 xv = *(const float4*)(xr + c);
    float4 w1v = *(const float4*)(w1 + c);
    float4 w2v = *(const float4*)(w2 + c);
    s1 += xv.x * w1v.x + xv.y * w1v.y + xv.z * w1v.z + xv.w * w1v.w;
    s2 += xv.x * w2v.x + xv.y * w2v.y + xv.z * w2v.z + xv.w * w2v.w;
  }
#pragma unroll
  for (int off = 16; off > 0; off >>= 1) {
    s1 += __shfl_xor(s1, off, 32);
    s2 += __shfl_xor(s2, off, 32);
  }
  if (lane == 0) { o1[row] = s1; o2[row] = s2; }
}

__global__ void init_node(float* __restrict__ m, float* __restrict__ den, int N) {
  int i = blockIdx.x * blockDim.x + threadIdx.x;
  if (i < N) { m[i] = -INFINITY; den[i] = 0.f; }
}

// agg[n*C+c] += bA[c] + bB[c]
__global__ void bias_add(float* __restrict__ agg, const float* __restrict__ bA,
                         const float* __restrict__ bB, int total, int Cmask) {
  for (int i = blockIdx.x * blockDim.x + threadIdx.x; i < total;
       i += gridDim.x * blockDim.x) {
    int c = i & Cmask;
    agg[i] += bA[c] + bB[c];
  }
}

__global__ void relu_ip(float* __restrict__ p, int total) {
  for (int i = blockIdx.x * blockDim.x + threadIdx.x; i < total;
       i += gridDim.x * blockDim.x) {
    float v = p[i];
    p[i] = v > 0.f ? v : 0.f;
  }
}

// float atomic max via signed-max / unsigned-min idiom (init = -inf)
__device__ __forceinline__ void atomicMaxF(float* addr, float v) {
  if (v >= 0.f) atomicMax((int*)addr, __float_as_int(v));
  else          atomicMin((unsigned int*)addr, __float_as_uint(v));
}

// pass 1: logits + segment max
__global__ void edge_logits_max(const long long* __restrict__ src,
                                const long long* __restrict__ dst,
                                const float* __restrict__ as,
                                const float* __restrict__ ad,
                                float* __restrict__ elog,
                                float* __restrict__ m, int E) {
  int i = blockIdx.x * blockDim.x + threadIdx.x;
  if (i >= E) return;
  int s = (int)src[i], d = (int)dst[i];
  float l = as[s] + ad[d];
  l = l > 0.f ? l : NEG_SLOPE * l;
  elog[i] = l;
  atomicMaxF(&m[d], l);
}

// pass 2: e = exp(l - m[dst]); den[dst] += e
__global__ void edge_exp_sum(const long long* __restrict__ dst,
                             const float* __restrict__ m,
                             float* __restrict__ elog,
                             float* __restrict__ den, int E) {
  int i = blockIdx.x * blockDim.x + threadIdx.x;
  if (i >= E) return;
  int d = (int)dst[i];
  float e = expf(elog[i] - m[d]);
  elog[i] = e;
  atomicAdd(&den[d], e);
}

// pass 3: alpha = e/(den+eps); agg[dst,:] += alpha * H[src,:]
// one wave per edge; b128/b64 gathers; lanes hit distinct atomic addresses.
__global__ void edge_scatter(const long long* __restrict__ src,
                             const long long* __restrict__ dst,
                             const float* __restrict__ elog,
                             const float* __restrict__ den,
                             const float* __restrict__ H,
                             float* __restrict__ agg,
                             float* __restrict__ alphaOut,
                             int E, int C) {
  int e    = blockIdx.x * (blockDim.x >> 5) + (threadIdx.x >> 5);
  int lane = threadIdx.x & 31;
  if (e >= E) return;
  int s = (int)src[e], d = (int)dst[e];
  float alpha = elog[e] / (den[d] + GAT_EPS);
  if (alphaOut && lane == 0) alphaOut[e] = alpha;
  const float* __restrict__ hr = H + (size_t)s * C;
  float* __restrict__ ar = agg + (size_t)d * C;
  if ((C & 127) == 0) {                       // C == 128: float4 per lane
    for (int c = lane * 4; c < C; c += 128) {
      float4 hv = *(const float4*)(hr + c);
      atomicAdd(&ar[c + 0], hv.x * alpha);
      atomicAdd(&ar[c + 1], hv.y * alpha);
      atomicAdd(&ar[c + 2], hv.z * alpha);
      atomicAdd(&ar[c + 3], hv.w * alpha);
    }
  } else if ((C & 63) == 0) {                 // C == 64: float2 per lane
    for (int c = lane * 2; c < C; c += 64) {
      float2 hv = *(const float2*)(hr + c);
      atomicAdd(&ar[c + 0], hv.x * alpha);
      atomicAdd(&ar[c + 1], hv.y * alpha);
    }
  } else {
    for (int c = lane; c < C; c += 32) atomicAdd(&ar[c], hr[c] * alpha);
  }
}

// ---------------------------------------------------------------------------
extern "C" void kernel_launch(void* const* d_in, const int* in_sizes, int n_in,
                              void* d_out, int out_size, void* d_ws, size_t ws_size,
                              hipStream_t stream) {
  const float*     x        = (const float*)d_in[0];
  const long long* ei       = (const long long*)d_in[1];  // int64 [2, E]
  const float*     W_src1   = (const float*)d_in[2];
  const float*     W_dst1   = (const float*)d_in[3];
  const float*     att_src1 = (const float*)d_in[4];
  const float*     att_dst1 = (const float*)d_in[5];
  const float*     b1       = (const float*)d_in[6];
  const float*     W_lin1   = (const float*)d_in[7];
  const float*     b_lin1   = (const float*)d_in[8];
  const float*     W_src2   = (const float*)d_in[9];
  const float*     W_dst2   = (const float*)d_in[10];
  const float*     att_src2 = (const float*)d_in[11];
  const float*     att_dst2 = (const float*)d_in[12];
  const float*     b2       = (const float*)d_in[13];
  const float*     W_lin2   = (const float*)d_in[14];
  const float*     b_lin2   = (const float*)d_in[15];

  const int N = in_sizes[0] / DIN;   // 50000
  const int E = in_sizes[1] / 2;     // 800000
  const long long* srcI = ei;
  const long long* dstI = ei + E;

  // ---- scratch layout (floats), ~17.1M floats ≈ 68.5 MB ----
  float* ws   = (float*)d_ws;
  float* h1s  = ws;  ws += (size_t)N * HID;   // x @ W_src1
  float* agg1 = ws;  ws += (size_t)N * HID;   // x @ W_lin1 -> +bias -> +msgs -> relu -> h
  float* h2s  = ws;  ws += (size_t)N * OUTC;  // h @ W_src2
  float* elog = ws;  ws += E;                 // logits -> exp(e)
  float* a_s  = ws;  ws += N;
  float* a_d  = ws;  ws += N;
  float* mbuf = ws;  ws += N;
  float* den  = ws;  ws += N;
  float* wv1  = ws;  ws += DIN;               // W_src @ att_src (folded)
  float* wv2  = ws;  ws += DIN;               // W_dst @ att_dst (folded)

  float* outAgg   = (float*)d_out;                       // [N, OUTC]
  float* alphaOut = (float*)d_out + (size_t)N * OUTC;    // [E]

  const dim3 blk128(128);
  const int  rowTiles     = N / 16;               // N % 16 == 0 (50000/16=3125)
  const int  gemmBlocksX  = (rowTiles + 3) / 4;   // 4 waves per block
  const int  edgeBlocks1  = (E + 255) / 256;
  const int  edgeWaves    = (E + 7) / 8;          // 8 waves per 256-thread block
  const int  nodeBlocks   = (N + 255) / 256;
  const int  rowBlocks    = (N + 7) / 8;

  // ================= layer 1 =================
  {
    dim3 g1(gemmBlocksX, HID / 64);
    gemm_wmma_f32<<<g1, blk128, 0, stream>>>(x, W_src1, h1s, N, DIN, HID);
    gemm_wmma_f32<<<g1, blk128, 0, stream>>>(x, W_lin1, agg1, N, DIN, HID);
  }
  wvec2<<<1, DIN, 0, stream>>>(W_src1, att_src1, W_dst1, att_dst1, wv1, wv2, DIN, HID);
  rowdot2<<<rowBlocks, 256, 0, stream>>>(x, wv1, wv2, a_s, a_d, N, DIN);
  init_node<<<nodeBlocks, 256, 0, stream>>>(mbuf, den, N);
  bias_add<<<2048, 256, 0, stream>>>(agg1, b1, b_lin1, N * HID, HID - 1);
  edge_logits_max<<<edgeBlocks1, 256, 0, stream>>>(srcI, dstI, a_s, a_d, elog, mbuf, E);
  edge_exp_sum<<<edgeBlocks1, 256, 0, stream>>>(dstI, mbuf, elog, den, E);
  edge_scatter<<<edgeWaves, 256, 0, stream>>>(srcI, dstI, elog, den, h1s, agg1,
                                              nullptr, E, HID);
  relu_ip<<<2048, 256, 0, stream>>>(agg1, N * HID);      // agg1 is now h

  // ================= layer 2 =================
  {
    dim3 g2(gemmBlocksX, 1);                             // OUTC == 64
    gemm_wmma_f32<<<g2, blk128, 0, stream>>>(agg1, W_src2, h2s, N, HID, OUTC);
    gemm_wmma_f32<<<g2, blk128, 0, stream>>>(agg1, W_lin2, outAgg, N, HID, OUTC);
  }
  wvec2<<<1, HID, 0, stream>>>(W_src2, att_src2, W_dst2, att_dst2, wv1, wv2, HID, OUTC);
  rowdot2<<<rowBlocks, 256, 0, stream>>>(agg1, wv1, wv2, a_s, a_d, N, HID);
  init_node<<<nodeBlocks, 256, 0, stream>>>(mbuf, den, N);
  bias_add<<<2048, 256, 0, stream>>>(outAgg, b2, b_lin2, N * OUTC, OUTC - 1);
  edge_logits_max<<<edgeBlocks1, 256, 0, stream>>>(srcI, dstI, a_s, a_d, elog, mbuf, E);
  edge_exp_sum<<<edgeBlocks1, 256, 0, stream>>>(dstI, mbuf, elog, den, E);
  edge_scatter<<<edgeWaves, 256, 0, stream>>>(srcI, dstI, elog, den, h2s, outAgg,
                                              alphaOut, E, OUTC);
}